// FAPELoss_26293789786381
// MI455X (gfx1250) — compile-verified
//
#include <hip/hip_runtime.h>
#include <math.h>

typedef __attribute__((ext_vector_type(2))) float v2f;
typedef __attribute__((ext_vector_type(8))) float v8f;

#define BD 2
#define ND 1024
#define AD 14
#define KPTS (ND*AD)          /* 14336 points per batch */
#define GROUPS (ND/4)         /* 256 groups of 4 frames (exact, no padding) */
#define KTILES (KPTS/16)      /* 896 tiles of 16 points */
#define NBLK (BD*GROUPS)      /* 512 blocks in main kernel */
#define EPSF 1e-8f
#define DCLAMP 10.0f

/* workspace layout (float offsets; 16B-aligned sections) */
#define OFF_M   0                       /* BD*ND*24 = 49152 : Mp(12)+Mt(12) per frame */
#define OFF_RES (OFF_M + BD*ND*24)      /* 49152 : residue masks (2048) */
#define OFF_PP  (OFF_RES + BD*ND)       /* 51200 : pred points float4 (x,y,z,1) */
#define OFF_PT  (OFF_PP + BD*KPTS*4)    /* 165888 : true points float4 */
#define OFF_CM  (OFF_PT + BD*KPTS*4)    /* 280576 : column mask as float */
#define OFF_PART (OFF_CM + BD*KPTS)     /* 309248 : per-block partial sums (512) */

__device__ inline void frame_axes(const float* bb, float e[3][3], float ca[3]) {
  // bb: 9 floats = N, CA, C atoms
  float nx=bb[0], ny=bb[1], nz=bb[2];
  float cax=bb[3], cay=bb[4], caz=bb[5];
  float cx=bb[6], cy=bb[7], cz=bb[8];
  ca[0]=cax; ca[1]=cay; ca[2]=caz;
  float v1x=cx-cax, v1y=cy-cay, v1z=cz-caz;
  float v2x=nx-cax, v2y=ny-cay, v2z=nz-caz;
  float i1 = 1.0f/sqrtf(v1x*v1x+v1y*v1y+v1z*v1z+EPSF);
  float e1x=v1x*i1, e1y=v1y*i1, e1z=v1z*i1;
  float d12 = e1x*v2x+e1y*v2y+e1z*v2z;
  float u2x=v2x-e1x*d12, u2y=v2y-e1y*d12, u2z=v2z-e1z*d12;
  float i2 = 1.0f/sqrtf(u2x*u2x+u2y*u2y+u2z*u2z+EPSF);
  float e2x=u2x*i2, e2y=u2y*i2, e2z=u2z*i2;
  float e3x=e1y*e2z-e1z*e2y, e3y=e1z*e2x-e1x*e2z, e3z=e1x*e2y-e1y*e2x;
  e[0][0]=e1x; e[0][1]=e1y; e[0][2]=e1z;
  e[1][0]=e2x; e[1][1]=e2y; e[1][2]=e2z;
  e[2][0]=e3x; e[2][1]=e3y; e[2][2]=e3z;
}

/* Stage 1: per-residue affine matrices Mp (pred) and Mt (true) + residue mask */
__global__ __launch_bounds__(256) void fape_frames(const float* __restrict__ pred,
                                                   const float* __restrict__ tru,
                                                   const int* __restrict__ cmask,
                                                   float* __restrict__ ws) {
  int idx = blockIdx.x*blockDim.x + threadIdx.x;
  if (idx >= BD*ND) return;
  float* M = ws + OFF_M + (size_t)idx*24;
  size_t rbase = (size_t)idx*AD;
  float ep[3][3], cap[3], et[3][3], cat[3];
  frame_axes(pred + rbase*3, ep, cap);
  frame_axes(tru  + rbase*3, et, cat);
  #pragma unroll
  for (int c=0;c<3;c++) {
    // Mp row c: (+pred axis c, -dot(axis, CA_p))  => +pred_local
    M[c*4+0]=ep[c][0]; M[c*4+1]=ep[c][1]; M[c*4+2]=ep[c][2];
    M[c*4+3]= -(ep[c][0]*cap[0]+ep[c][1]*cap[1]+ep[c][2]*cap[2]);
    // Mt row c: (-true axis c, +dot(axis, CA_t)) => -true_local
    M[12+c*4+0]=-et[c][0]; M[12+c*4+1]=-et[c][1]; M[12+c*4+2]=-et[c][2];
    M[12+c*4+3]= (et[c][0]*cat[0]+et[c][1]*cat[1]+et[c][2]*cat[2]);
  }
  int any = 0;
  #pragma unroll
  for (int j=0;j<AD;j++) any |= cmask[rbase + j];
  ws[OFF_RES + idx] = any ? 1.f : 0.f;
}

/* Stage 1b: homogeneous float4 points + column masks */
__global__ __launch_bounds__(256) void fape_pack(const float* __restrict__ pred,
                                                 const float* __restrict__ tru,
                                                 const int* __restrict__ cmask,
                                                 float* __restrict__ ws) {
  int idx = blockIdx.x*blockDim.x + threadIdx.x;
  if (idx >= BD*KPTS) return;
  float* pp = ws + OFF_PP + (size_t)idx*4;
  float* pt = ws + OFF_PT + (size_t)idx*4;
  size_t s = (size_t)idx*3;
  pp[0]=pred[s]; pp[1]=pred[s+1]; pp[2]=pred[s+2]; pp[3]=1.f;
  pt[0]=tru[s];  pt[1]=tru[s+1];  pt[2]=tru[s+2];  pt[3]=1.f;
  ws[OFF_CM + idx] = cmask[idx] ? 1.f : 0.f;
}

/* Stage 2: main FAPE via chained V_WMMA_F32_16X16X4_F32.
   A tile: rows 0-2,3-5 = frames 4g+0,4g+1 (lane half 0 of D);
           rows 8-10,11-13 = frames 4g+2,4g+3 (lane half 1 of D);
           rows 6,7,14,15 zero.
   Both halves then run identical branchless epilogue code. */
__global__ __launch_bounds__(128) void fape_wmma(const float* __restrict__ ws,
                                                 float* __restrict__ partials) {
  int bi = blockIdx.x / GROUPS;
  int g  = blockIdx.x % GROUPS;
  int tid = threadIdx.x, lane = tid & 31, wave = tid >> 5;
  int r = lane & 15;              // A-matrix row (M) held by this lane
  bool hi = (lane >= 16);         // high half: A holds K2,K3; D holds rows 8..15
  int col = lane & 15;            // D-matrix column (N)

  // A operands: row r -> (frame, component); rows 6,7,14,15 stay zero
  v2f Ap = {0.f, 0.f}, At = {0.f, 0.f};
  bool valid = (r < 6) | (r >= 8 && r < 14);
  if (valid) {
    int rv = (r >= 8) ? (r - 2) : r;     // 0..11 over 4 frames x 3 comps
    int f = rv / 3, c = rv - 3*f;
    int frame = 4*g + f;
    const float* mp = ws + OFF_M + ((size_t)(bi*ND + frame))*24 + c*4 + (hi ? 2 : 0);
    Ap[0]=mp[0];  Ap[1]=mp[1];
    At[0]=mp[12]; At[1]=mp[13];
  }

  // residue masks: half0 lanes handle frames 4g+0,4g+1; half1 handle 4g+2,4g+3
  const float* rmp = ws + OFF_RES + (size_t)bi*ND + 4*g;
  float mA = hi ? rmp[2] : rmp[0];
  float mB = hi ? rmp[3] : rmp[1];

  const float* PpB = ws + OFF_PP + (size_t)bi*KPTS*4;
  const float* PtB = ws + OFF_PT + (size_t)bi*KPTS*4;
  const float* cmB = ws + OFF_CM + (size_t)bi*KPTS;
  int sub = hi ? 2 : 0;

  float acc = 0.f;
  for (int t = wave; t < KTILES; t += 4) {
    int p = t*16 + col;
    v2f Bp = { PpB[(size_t)p*4 + sub], PpB[(size_t)p*4 + sub + 1] };
    v2f Bt = { PtB[(size_t)p*4 + sub], PtB[(size_t)p*4 + sub + 1] };
    float mk = cmB[p];

    v8f d = {0.f,0.f,0.f,0.f,0.f,0.f,0.f,0.f};
    // d  = Mt x (true,1)
    d = __builtin_amdgcn_wmma_f32_16x16x4_f32(false, At, false, Bt, (short)0, d, false, false);
    // d += Mp x (pred,1)  -> pred_local - true_local per (row, point)
    d = __builtin_amdgcn_wmma_f32_16x16x4_f32(false, Ap, false, Bp, (short)0, d, false, false);

    // rows (0,1,2)/(8,9,10) -> first frame of this half; (3,4,5)/(11,12,13) -> second
    float sA = d[0]*d[0] + d[1]*d[1] + d[2]*d[2];
    float sB = d[3]*d[3] + d[4]*d[4] + d[5]*d[5];
    float cA = fminf(__builtin_amdgcn_sqrtf(sA + EPSF), DCLAMP);
    float cB = fminf(__builtin_amdgcn_sqrtf(sB + EPSF), DCLAMP);
    acc += mk * (mA*cA + mB*cB);
  }

  // deterministic wave + block reduction
  #pragma unroll
  for (int off = 16; off >= 1; off >>= 1) acc += __shfl_xor(acc, off, 32);
  __shared__ float ssum[4];
  if (lane == 0) ssum[wave] = acc;
  __syncthreads();
  if (tid == 0) partials[blockIdx.x] = (ssum[0] + ssum[1]) + (ssum[2] + ssum[3]);
}

/* Stage 3: fixed-order final reduction + scaling */
__global__ __launch_bounds__(512) void fape_final(const float* __restrict__ partials,
                                                  float* __restrict__ out) {
  __shared__ float sm[512];
  float s = 0.f;
  for (int i = threadIdx.x; i < NBLK; i += 512) s += partials[i];
  sm[threadIdx.x] = s;
  __syncthreads();
  for (int st = 256; st >= 1; st >>= 1) {
    if (threadIdx.x < st) sm[threadIdx.x] += sm[threadIdx.x + st];
    __syncthreads();
  }
  if (threadIdx.x == 0) {
    // mean over b*n*(n*a) = 29,360,128 elements, then / SCALE(10)
    out[0] = sm[0] * (1.0f / 293601280.0f);
  }
}

extern "C" void kernel_launch(void* const* d_in, const int* in_sizes, int n_in,
                              void* d_out, int out_size, void* d_ws, size_t ws_size,
                              hipStream_t stream) {
  const float* pred = (const float*)d_in[0];
  const float* tru  = (const float*)d_in[1];
  const int*   cm   = (const int*)d_in[2];
  float* ws  = (float*)d_ws;
  float* out = (float*)d_out;

  fape_frames<<<(BD*ND + 255)/256, 256, 0, stream>>>(pred, tru, cm, ws);
  fape_pack  <<<(BD*KPTS + 255)/256, 256, 0, stream>>>(pred, tru, cm, ws);
  fape_wmma  <<<NBLK, 128, 0, stream>>>(ws, ws + OFF_PART);
  fape_final <<<1, 512, 0, stream>>>(ws + OFF_PART, out);
}